// HCCF_Encoder_48619029790850
// MI455X (gfx1250) — compile-verified
//
#include <hip/hip_runtime.h>
#include <stdint.h>
#include <stddef.h>

#define EMB        64
#define EMB4       (EMB / 4)
#define EPB        128    // edges per block in the SpMM kernel
#define TPB        256
#define N_LAYERS   3
#define LAYER_INV  (1.0f / (float)(N_LAYERS + 1))

// ---------------------------------------------------------------------------
// CDNA5 helpers (gfx1250)
// ---------------------------------------------------------------------------

// LDS byte offset of a shared-memory pointer (generic LDS aperture: low 32
// bits of the flat address are the workgroup-relative LDS offset).
__device__ __forceinline__ uint32_t lds_off_u32(const void* p) {
  return (uint32_t)(uintptr_t)p;
}

// Async copy of one b32 per lane from global memory into LDS (ASYNCcnt path).
__device__ __forceinline__ void async_g2l_b32(uint32_t lds_off, const int* gaddr) {
  asm volatile("global_load_async_to_lds_b32 %0, %1, off"
               :: "v"(lds_off), "v"(gaddr)
               : "memory");
}

__device__ __forceinline__ void wait_async0() {
  asm volatile("s_wait_asynccnt 0" ::: "memory");
}

// Hardware fp32 atomic add, device scope (coherent across WGPs in L2),
// no return value (STOREcnt, fire-and-forget).
__device__ __forceinline__ void atomic_add_f32_dev(float* p, float v) {
  asm volatile("global_atomic_add_f32 %0, %1, off scope:SCOPE_DEV"
               :: "v"(p), "v"(v)
               : "memory");
}

// ---------------------------------------------------------------------------
// Kernels
// ---------------------------------------------------------------------------

__global__ void k_zero_u32(unsigned* __restrict__ p, int n) {
  int i = blockIdx.x * blockDim.x + threadIdx.x;
  if (i < n) p[i] = 0u;
}

__global__ void k_fill0_f4(float4* __restrict__ p, int n4) {
  int i = blockIdx.x * blockDim.x + threadIdx.x;
  if (i < n4) p[i] = make_float4(0.f, 0.f, 0.f, 0.f);
}

// state[i] = src[i];  sum[i] = src[i] * LAYER_INV   (layer-0 term pre-scaled)
__global__ void k_init_f4(const float4* __restrict__ src,
                          float4* __restrict__ state,
                          float4* __restrict__ sum, int n4) {
  int i = blockIdx.x * blockDim.x + threadIdx.x;
  if (i < n4) {
    float4 v = src[i];
    state[i] = v;
    float4 w = v;
    w.x *= LAYER_INV; w.y *= LAYER_INV; w.z *= LAYER_INV; w.w *= LAYER_INV;
    sum[i] = w;
  }
}

// Degree histogram over the edge list.
__global__ void k_degrees(const int* __restrict__ u_idx, const int* __restrict__ i_idx,
                          unsigned* __restrict__ deg_u, unsigned* __restrict__ deg_i,
                          int n_edges) {
  int e = blockIdx.x * blockDim.x + threadIdx.x;
  if (e < n_edges) {
    atomicAdd(&deg_u[u_idx[e]], 1u);
    atomicAdd(&deg_i[i_idx[e]], 1u);
  }
}

// inv = 1/max(d,1), isq = max(d,1)^-0.5
__global__ void k_norms(const unsigned* __restrict__ deg,
                        float* __restrict__ inv, float* __restrict__ isq, int n) {
  int i = blockIdx.x * blockDim.x + threadIdx.x;
  if (i < n) {
    unsigned du = deg[i];
    float d = (du == 0u) ? 1.0f : (float)du;
    inv[i] = 1.0f / d;
    isq[i] = 1.0f / sqrtf(d);
  }
}

// y[dst[e]] += x[src[e]] * xscale[src[e]]  for all edges.
// 16 lanes per edge, each lane one float4 (b128 gather + 4 f32 atomics).
// Edge indices for the whole block are staged into LDS asynchronously.
__global__ __launch_bounds__(TPB) void k_spmm_edges(
    const int* __restrict__ src_idx, const int* __restrict__ dst_idx,
    const float* __restrict__ x, const float* __restrict__ xscale,
    float* __restrict__ y, int n_edges) {
  __shared__ int s_idx[2 * EPB];   // [0..EPB) = src, [EPB..2*EPB) = dst

  const int t    = threadIdx.x;
  const int base = blockIdx.x * EPB;

  if (base + EPB <= n_edges) {
    // Full block: all 256 lanes issue one async b32 (128 src + 128 dst).
    const int* g = (t < EPB) ? (src_idx + base + t)
                             : (dst_idx + base + (t - EPB));
    async_g2l_b32(lds_off_u32(&s_idx[t]), g);
    wait_async0();                 // per-wave ASYNCcnt drain
  } else {
    // Tail block: guarded synchronous loads.
    if (t < EPB) {
      int j = base + t;
      s_idx[t]       = (j < n_edges) ? src_idx[j] : 0;
      s_idx[EPB + t] = (j < n_edges) ? dst_idx[j] : 0;
    }
  }
  __syncthreads();

  const int lane16 = t & 15;       // float4 slot within the 64-wide row
  const int egrp   = t >> 4;       // 16 edges processed per pass
#pragma unroll
  for (int p = 0; p < EPB / 16; ++p) {
    const int eb = p * 16 + egrp;
    if (base + eb < n_edges) {
      const int s = s_idx[eb];
      const int d = s_idx[EPB + eb];
      const float sc = xscale[s];
      const float4 v = ((const float4*)(x + (size_t)s * EMB))[lane16];
      float* yp = y + (size_t)d * EMB + lane16 * 4;
      atomic_add_f32_dev(yp + 0, v.x * sc);
      atomic_add_f32_dev(yp + 1, v.y * sc);
      atomic_add_f32_dev(yp + 2, v.z * sc);
      atomic_add_f32_dev(yp + 3, v.w * sc);
    }
  }
}

// e_out[n,:] = acc[n,:] * scale[n];  sum_out[n,:] += e_out[n,:] * LAYER_INV
__global__ void k_node_update(const float* __restrict__ acc,
                              const float* __restrict__ scale,
                              float* __restrict__ e_out,
                              float* __restrict__ sum_out, int n_nodes) {
  int i  = blockIdx.x * blockDim.x + threadIdx.x;  // over n_nodes * EMB4
  int n4 = n_nodes * EMB4;
  if (i < n4) {
    int node = i >> 4;             // EMB4 == 16
    float s = scale[node];
    float4 v = ((const float4*)acc)[i];
    v.x *= s; v.y *= s; v.z *= s; v.w *= s;
    ((float4*)e_out)[i] = v;
    float4 o = ((const float4*)sum_out)[i];
    o.x += v.x * LAYER_INV; o.y += v.y * LAYER_INV;
    o.z += v.z * LAYER_INV; o.w += v.w * LAYER_INV;
    ((float4*)sum_out)[i] = o;
  }
}

// ---------------------------------------------------------------------------
// Launch
// ---------------------------------------------------------------------------

static inline int cdiv_i(long a, long b) { return (int)((a + b - 1) / b); }

extern "C" void kernel_launch(void* const* d_in, const int* in_sizes, int n_in,
                              void* d_out, int out_size, void* d_ws, size_t ws_size,
                              hipStream_t stream) {
  const float* user_emb = (const float*)d_in[0];
  const float* item_emb = (const float*)d_in[1];
  const int*   u_idx    = (const int*)d_in[2];
  const int*   i_idx    = (const int*)d_in[3];

  const int U = in_sizes[0] / EMB;
  const int I = in_sizes[1] / EMB;
  const int E = in_sizes[2];

  // Workspace layout (floats / u32s share the arena).
  float*    ws     = (float*)d_ws;
  unsigned* deg_u  = (unsigned*)ws;                  // U
  unsigned* deg_i  = deg_u + U;                      // I
  float*    du_inv = ws + (size_t)(U + I);           // U
  float*    du_is  = du_inv + U;                     // U
  float*    di_inv = du_is + U;                      // I
  float*    di_is  = di_inv + I;                     // I
  float*    user_e = di_is + I;                      // U*EMB
  float*    item_e = user_e + (size_t)U * EMB;       // I*EMB
  float*    acc_u  = item_e + (size_t)I * EMB;       // U*EMB
  float*    acc_i  = acc_u + (size_t)U * EMB;        // I*EMB  (contiguous after acc_u)

  float* out_u = (float*)d_out;                      // sum_u * LAYER_INV
  float* out_i = out_u + (size_t)U * EMB;            // sum_i * LAYER_INV

  const int U4 = U * EMB4, I4 = I * EMB4;
  const int UI4 = U4 + I4;                           // acc_u ++ acc_i region
  const int sgrid = cdiv_i(E, EPB);

  // Degrees + norms.
  k_zero_u32<<<cdiv_i(U + I, 256), 256, 0, stream>>>(deg_u, U + I);
  k_degrees<<<cdiv_i(E, 256), 256, 0, stream>>>(u_idx, i_idx, deg_u, deg_i, E);
  k_norms<<<cdiv_i(U, 256), 256, 0, stream>>>(deg_u, du_inv, du_is, U);
  k_norms<<<cdiv_i(I, 256), 256, 0, stream>>>(deg_i, di_inv, di_is, I);

  // user_e = user_emb, out_u = user_emb * LAYER_INV ; items likewise.
  k_init_f4<<<cdiv_i(U4, 256), 256, 0, stream>>>(
      (const float4*)user_emb, (float4*)user_e, (float4*)out_u, U4);
  k_init_f4<<<cdiv_i(I4, 256), 256, 0, stream>>>(
      (const float4*)item_emb, (float4*)item_e, (float4*)out_i, I4);

  for (int l = 0; l < N_LAYERS; ++l) {
    // ---- user propagation: u -> items -> u ----
    // Zero both accumulators in one contiguous pass (spmm #1 accumulates into acc_i and does not touch acc_u).
    k_fill0_f4<<<cdiv_i(UI4, 256), 256, 0, stream>>>((float4*)acc_u, UI4);
    // acc_i = R^T (user_e * du_is)
    k_spmm_edges<<<sgrid, TPB, 0, stream>>>(u_idx, i_idx, user_e, du_is, acc_i, E);
    // acc_u = R (acc_i * di_inv)
    k_spmm_edges<<<sgrid, TPB, 0, stream>>>(i_idx, u_idx, acc_i, di_inv, acc_u, E);
    // user_e = acc_u * du_is ; out_u += user_e * LAYER_INV
    k_node_update<<<cdiv_i(U4, 256), 256, 0, stream>>>(acc_u, du_is, user_e, out_u, U);

    // ---- item propagation: i -> users -> i ----
    k_fill0_f4<<<cdiv_i(UI4, 256), 256, 0, stream>>>((float4*)acc_u, UI4);
    // acc_u = R (item_e * di_is)
    k_spmm_edges<<<sgrid, TPB, 0, stream>>>(i_idx, u_idx, item_e, di_is, acc_u, E);
    // acc_i = R^T (acc_u * du_inv)
    k_spmm_edges<<<sgrid, TPB, 0, stream>>>(u_idx, i_idx, acc_u, du_inv, acc_i, E);
    // item_e = acc_i * di_is ; out_i += item_e * LAYER_INV
    k_node_update<<<cdiv_i(I4, 256), 256, 0, stream>>>(acc_i, di_is, item_e, out_i, I);
  }
  // Final averaging already folded into k_init_f4 / k_node_update.
}